// TGNLayerGraphSumEmbedding_48747878810094
// MI455X (gfx1250) — compile-verified
//
#include <hip/hip_runtime.h>

typedef __attribute__((ext_vector_type(2))) float v2f;
typedef __attribute__((ext_vector_type(4))) float v4f;
typedef __attribute__((ext_vector_type(8))) float v8f;

#define N_NODES 50000
#define KNBR    16
#define DIM     256
#define EDIM    100
#define TDIM    100
#define ET      (EDIM + TDIM)       // 200
#define NTILES  (N_NODES / 16)      // 3125 (exact)
#define NB2     ((NTILES + 1) / 2)  // 1563 blocks, 2 row-tiles each (last clamps)

// Packed-weight layout: for K-group g and column n, 4 consecutive floats
//   pack[(g*256 + n)*4 + j] = W[4g + j][n]
// Lane B-fragment {W[kb][n], W[kb+1][n]} (kb = 4g + khalf) = ONE global_load_b64.
#define G_W1D 64                    // W1 rows   0..255 -> groups 0..63
#define G_WET 50                    // W1 rows 256..455 -> groups 0..49
#define G_W2  128                   // W2 rows   0..511 -> groups 0..127
#define PACK_TOTAL ((G_W1D + G_WET + G_W2) * DIM * 4)   // 247808 floats

#define WMMA_F32(a, b, c) \
  __builtin_amdgcn_wmma_f32_16x16x4_f32(false, (a), false, (b), (short)0, (c), false, false)

__global__ __launch_bounds__(256) void k_pack_weights(
    const float* __restrict__ W1, const float* __restrict__ W2,
    float* __restrict__ pW1d, float* __restrict__ pWet, float* __restrict__ pW2)
{
  const int idx = blockIdx.x * 256 + threadIdx.x;
  if (idx >= PACK_TOTAL) return;
  const int j = idx & 3;
  const int n = (idx >> 2) & (DIM - 1);
  const int g = idx >> 10;
  if (g < G_W1D) {
    pW1d[idx] = W1[(size_t)(g * 4 + j) * DIM + n];
  } else if (g < G_W1D + G_WET) {
    const int gl = g - G_W1D;
    pWet[(size_t)gl * 1024 + (n << 2) + j] = W1[(size_t)(DIM + gl * 4 + j) * DIM + n];
  } else {
    const int gl = g - (G_W1D + G_WET);
    pW2[(size_t)gl * 1024 + (n << 2) + j] = W2[(size_t)(gl * 4 + j) * DIM + n];
  }
}

// -------------------------------------------------------------------------
// Kernel 0: c_et[n,:] = b1 + sum_k(edge[n,k,:]) @ W1[256:356,:]
//                          + sum_k(time[n,k,:]) @ W1[356:456,:]
// -------------------------------------------------------------------------
#define ET_LD 204   // 204 % 64 = 12 -> 16 rows map to 16 distinct bank groups

__global__ __launch_bounds__(256) void k_precompute_et(
    const float* __restrict__ edge, const float* __restrict__ timef,
    const float* __restrict__ pWet, const float* __restrict__ b1,
    float* __restrict__ c_et)
{
  __shared__ float s_et[16][ET_LD];
  const int tid  = threadIdx.x;
  const int base = blockIdx.x * 16;

  if (tid < ET) {
    const bool isE = tid < EDIM;
    const float* src = isE ? edge : timef;
    const int col = isE ? tid : tid - EDIM;
    for (int r = 0; r < 16; ++r) {
      float acc = 0.f;
      const size_t rowbase = (size_t)(base + r) * KNBR;
      #pragma unroll
      for (int k = 0; k < KNBR; ++k)
        acc += src[(rowbase + k) * EDIM + col];
      s_et[r][tid] = acc;
    }
  }
  __syncthreads();

  const int lane  = tid & 31;
  const int wave  = tid >> 5;
  const int m     = lane & 15;          // A-row / N-col lane index
  const int khalf = (lane >> 4) * 2;    // K sub-pair select (0 or 2)
  const int mo    = (lane >> 4) * 8;    // C/D row offset for upper lane half

  const int ncol0 = wave * 32 + m;
  const int ncol1 = ncol0 + 16;
  const float* wb0 = pWet + (ncol0 << 2) + khalf;   // ncol1 pack = wb0 + 64

  v8f c0 = {}, c1 = {};
  #pragma unroll 5
  for (int kk = 0; kk < ET; kk += 4) {
    const int kb = kk + khalf;
    v2f a; a.x = s_et[m][kb]; a.y = s_et[m][kb + 1];
    const size_t go = (size_t)(kk >> 2) << 10;
    v2f b0 = *(const v2f*)(wb0 + go);
    v2f b1 = *(const v2f*)(wb0 + go + 64);
    c0 = WMMA_F32(a, b0, c0);
    c1 = WMMA_F32(a, b1, c1);
  }
  const float bias0 = b1[ncol0];
  const float bias1 = b1[ncol1];
  #pragma unroll
  for (int i = 0; i < 8; ++i) {
    const size_t row = (size_t)(base + i + mo) * DIM;
    c_et[row + ncol0] = c0[i] + bias0;
    c_et[row + ncol1] = c1[i] + bias1;
  }
}

// -------------------------------------------------------------------------
// Kernel 1 (run once per layer), TWO 16-row tiles per block (2x2 WMMA grid):
//   s_nbr = sum_k in_feat[nbr[n,k]]                      (gather, L2-resident)
//   agg   = relu(s_nbr @ W1[0:256,:] + c_et)
//   out   = in_feat[n,:] @ W2[0:256,:] + agg @ W2[256:512,:] + b2
// -------------------------------------------------------------------------
#define LDP 260     // 260 % 64 = 4 -> 16 rows hit 16 distinct banks; 1040B = 65*16

__global__ __launch_bounds__(256) void k_tgn_layer(
    const float* __restrict__ in_feat, const int* __restrict__ nbrs,
    const float* __restrict__ pW1d, const float* __restrict__ pW2,
    const float* __restrict__ b2, const float* __restrict__ c_et,
    float* __restrict__ out)
{
  __shared__ int   s_idx[32][16];
  __shared__ float s_nbr [32][LDP];
  __shared__ float s_self[32][LDP];
  __shared__ float s_cet [32][LDP];
  __shared__ float s_agg [32][LDP];

  const int tid   = threadIdx.x;
  const int t0    = blockIdx.x * 2;                       // first row tile
  const int t1    = t0 + 1;
  const bool valid1 = (t1 < NTILES);
  const int t1c   = valid1 ? t1 : (NTILES - 1);           // clamp for last block
  const int base0 = t0 * 16;
  const int base1 = t1c * 16;

  {
    const int r = tid >> 4, k = tid & 15;
    s_idx[r     ][k] = nbrs[(size_t)(base0 + r) * KNBR + k];
    s_idx[16 + r][k] = nbrs[(size_t)(base1 + r) * KNBR + k];
  }
  __syncthreads();

  // Gather-sum + stage self features and c_et, vectorized float4:
  // thread owns dims [d4, d4+3] of 8 rows -> global_load_b128 / ds_store_b128.
  {
    const int r0 = tid >> 6;          // 0..3
    const int d4 = (tid & 63) << 2;   // 0,4,...,252
    #pragma unroll
    for (int rb = 0; rb < 8; ++rb) {
      const int r = rb * 4 + r0;                          // 0..31
      const int nodebase = (r < 16) ? base0 + r : base1 + (r - 16);
      v4f acc = {};
      #pragma unroll
      for (int k = 0; k < KNBR; ++k)
        acc += *(const v4f*)(in_feat + (size_t)s_idx[r][k] * DIM + d4);
      *(v4f*)&s_nbr [r][d4] = acc;
      *(v4f*)&s_self[r][d4] = *(const v4f*)(in_feat + (size_t)nodebase * DIM + d4);
      *(v4f*)&s_cet [r][d4] = *(const v4f*)(c_et    + (size_t)nodebase * DIM + d4);
    }
  }
  __syncthreads();

  const int lane  = tid & 31;
  const int wave  = tid >> 5;
  const int m     = lane & 15;
  const int khalf = (lane >> 4) * 2;
  const int mo    = (lane >> 4) * 8;

  const int ncol0 = wave * 32 + m;
  const int ncol1 = ncol0 + 16;

  // ---- Phase A: agg = relu(s_nbr @ W1[0:256] + c_et) -> LDS ----
  {
    const float* wb0 = pW1d + (ncol0 << 2) + khalf;       // ncol1 = +64 floats
    v8f c00 = {}, c01 = {}, c10 = {}, c11 = {};
    #pragma unroll 8
    for (int kk = 0; kk < DIM; kk += 4) {
      const int kb = kk + khalf;
      v2f a0; a0.x = s_nbr[m     ][kb]; a0.y = s_nbr[m     ][kb + 1];
      v2f a1; a1.x = s_nbr[16 + m][kb]; a1.y = s_nbr[16 + m][kb + 1];
      const size_t go = (size_t)(kk >> 2) << 10;
      v2f b0 = *(const v2f*)(wb0 + go);
      v2f b1 = *(const v2f*)(wb0 + go + 64);
      c00 = WMMA_F32(a0, b0, c00);
      c01 = WMMA_F32(a0, b1, c01);
      c10 = WMMA_F32(a1, b0, c10);
      c11 = WMMA_F32(a1, b1, c11);
    }
    #pragma unroll
    for (int i = 0; i < 8; ++i) {
      const int r = i + mo;
      float v00 = c00[i] + s_cet[r     ][ncol0];
      float v01 = c01[i] + s_cet[r     ][ncol1];
      float v10 = c10[i] + s_cet[16 + r][ncol0];
      float v11 = c11[i] + s_cet[16 + r][ncol1];
      s_agg[r     ][ncol0] = v00 > 0.f ? v00 : 0.f;
      s_agg[r     ][ncol1] = v01 > 0.f ? v01 : 0.f;
      s_agg[16 + r][ncol0] = v10 > 0.f ? v10 : 0.f;
      s_agg[16 + r][ncol1] = v11 > 0.f ? v11 : 0.f;
    }
  }
  __syncthreads();

  // ---- Phase B: out = s_self @ W2[0:256] + s_agg @ W2[256:512] + b2 ----
  {
    const float* wa0 = pW2 + (ncol0 << 2) + khalf;                        // groups 0..63
    const float* wb0 = pW2 + ((size_t)64 << 10) + (ncol0 << 2) + khalf;   // groups 64..127
    v8f c00 = {}, c01 = {}, c10 = {}, c11 = {};
    #pragma unroll 8
    for (int kk = 0; kk < DIM; kk += 4) {
      const int kb = kk + khalf;
      v2f a0; a0.x = s_self[m     ][kb]; a0.y = s_self[m     ][kb + 1];
      v2f a1; a1.x = s_self[16 + m][kb]; a1.y = s_self[16 + m][kb + 1];
      const size_t go = (size_t)(kk >> 2) << 10;
      v2f b0 = *(const v2f*)(wa0 + go);
      v2f b1 = *(const v2f*)(wa0 + go + 64);
      c00 = WMMA_F32(a0, b0, c00);
      c01 = WMMA_F32(a0, b1, c01);
      c10 = WMMA_F32(a1, b0, c10);
      c11 = WMMA_F32(a1, b1, c11);
    }
    #pragma unroll 8
    for (int kk = 0; kk < DIM; kk += 4) {
      const int kb = kk + khalf;
      v2f a0; a0.x = s_agg[m     ][kb]; a0.y = s_agg[m     ][kb + 1];
      v2f a1; a1.x = s_agg[16 + m][kb]; a1.y = s_agg[16 + m][kb + 1];
      const size_t go = (size_t)(kk >> 2) << 10;
      v2f b0 = *(const v2f*)(wb0 + go);
      v2f b1 = *(const v2f*)(wb0 + go + 64);
      c00 = WMMA_F32(a0, b0, c00);
      c01 = WMMA_F32(a0, b1, c01);
      c10 = WMMA_F32(a1, b0, c10);
      c11 = WMMA_F32(a1, b1, c11);
    }
    const float bias0 = b2[ncol0];
    const float bias1 = b2[ncol1];
    #pragma unroll
    for (int i = 0; i < 8; ++i) {
      const size_t row = (size_t)(base0 + i + mo) * DIM;
      out[row + ncol0] = c00[i] + bias0;
      out[row + ncol1] = c01[i] + bias1;
    }
    if (valid1) {   // uniform branch; all WMMAs already done (EXEC all-1s there)
      #pragma unroll
      for (int i = 0; i < 8; ++i) {
        const size_t row = (size_t)(base1 + i + mo) * DIM;
        out[row + ncol0] = c10[i] + bias0;
        out[row + ncol1] = c11[i] + bias1;
      }
    }
  }
}

// -------------------------------------------------------------------------
extern "C" void kernel_launch(void* const* d_in, const int* in_sizes, int n_in,
                              void* d_out, int out_size, void* d_ws, size_t ws_size,
                              hipStream_t stream) {
  (void)in_sizes; (void)n_in; (void)out_size; (void)ws_size;
  const float* features  = (const float*)d_in[0];
  const int*   neighbors = (const int*)  d_in[1];
  const float* edge      = (const float*)d_in[2];
  const float* timef     = (const float*)d_in[3];
  const float* W1        = (const float*)d_in[4];
  const float* b1        = (const float*)d_in[5];
  const float* W2        = (const float*)d_in[6];
  const float* b2        = (const float*)d_in[7];
  float* out  = (float*)d_out;

  float* c_et = (float*)d_ws;                         // N*256 f32 (51.2 MB)
  float* out1 = c_et + (size_t)N_NODES * DIM;         // N*256 f32 (layer-1 result)
  float* pW1d = out1 + (size_t)N_NODES * DIM;         // 64 groups  (256 KB)
  float* pWet = pW1d + (size_t)G_W1D * DIM * 4;       // 50 groups  (200 KB)
  float* pW2  = pWet + (size_t)G_WET * DIM * 4;       // 128 groups (512 KB)

  k_pack_weights <<<(PACK_TOTAL + 255) / 256, 256, 0, stream>>>(W1, W2, pW1d, pWet, pW2);
  k_precompute_et<<<NTILES, 256, 0, stream>>>(edge, timef, pWet, b1, c_et);
  k_tgn_layer    <<<NB2, 256, 0, stream>>>(features, neighbors, pW1d, pW2, b2, c_et, out1);
  k_tgn_layer    <<<NB2, 256, 0, stream>>>(out1,     neighbors, pW1d, pW2, b2, c_et, out);
}